// GATModel_51908974739987
// MI455X (gfx1250) — compile-verified
//
#include <hip/hip_runtime.h>
#include <hip/hip_bf16.h>
#include <limits.h>

// ---------------------------------------------------------------------------
// GAT forward: hs = x_s@W_src (WMMA f32), segment softmax over dst, scatter
// aggregate, mean-pool per graph, final fc.  Sizes fixed by the reference.
// ---------------------------------------------------------------------------
#define NN      100000
#define EE      1600000
#define DIN     128
#define HH      2
#define CC      50
#define HC      100       // H*C
#define BB      64
#define NEG     0.2f

#define CT      7         // column tiles: ceil(100/16)
#define CPAD    112       // padded column count (7*16)
#define KSTRIDE 132       // LDS dword stride per column; 132 % 64 == 4 -> no bank conflicts

typedef float v2f __attribute__((ext_vector_type(2)));
typedef float v8f __attribute__((ext_vector_type(8)));

// ---- ordered-int encoding so integer atomicMax == float max ---------------
__device__ __forceinline__ int f2key(float f) {
    int i = __float_as_int(f);
    return (i >= 0) ? i : (i ^ 0x7FFFFFFF);
}
__device__ __forceinline__ float key2f(int k) {
    int i = (k >= 0) ? k : (k ^ 0x7FFFFFFF);
    return __int_as_float(i);
}

// ---------------------------------------------------------------------------
// K0: w_eff[d][h] = sum_c W_dst[d, h*C+c] * att_dst[h, c]      (128 x 2)
// ---------------------------------------------------------------------------
__global__ void k_prep(const float* __restrict__ W_dst,
                       const float* __restrict__ att_dst,
                       float* __restrict__ weff) {
    int t = threadIdx.x;               // one block of 256
    if (t >= DIN * HH) return;
    int d = t >> 1, h = t & 1;
    float acc = 0.f;
    #pragma unroll 10
    for (int c = 0; c < CC; ++c)
        acc += W_dst[d * HC + h * CC + c] * att_dst[h * CC + c];
    weff[d * HH + h] = acc;
}

// ---------------------------------------------------------------------------
// K1: init scratch (out_feat=0, mI=INT_MIN, denom=0, pooled=0, counts=0)
// ---------------------------------------------------------------------------
__global__ void k_init(float* __restrict__ outf, int* __restrict__ mI,
                       float* __restrict__ denom, float* __restrict__ pooled,
                       float* __restrict__ counts) {
    long long t = (long long)blockIdx.x * blockDim.x + threadIdx.x;
    if (t < (long long)NN * HC) outf[t] = 0.f;
    if (t < NN * HH) { mI[t] = INT_MIN; denom[t] = 0.f; }
    if (t < BB * HC) pooled[t] = 0.f;
    if (t < BB)      counts[t] = 0.f;
}

// ---------------------------------------------------------------------------
// K2: hs[N x 100] = X[N x 128] * W[128 x 100] via V_WMMA_F32_16X16X4_F32.
// W is transposed + zero-padded into LDS (Bs[col][k], stride 132 dwords, so
// the 64 dwords of a wave's b64 fragment hit 64 distinct banks).  Each wave
// owns a 16-row strip and all 7 column tiles: per k-step, 1 global b64 (A),
// 7 ds b64 (B), 7 WMMA.  Zero-padded columns 100..111 are exact (contribute
// nothing); output store masks them off.
// ---------------------------------------------------------------------------
__global__ void k_gemm_wmma(const float* __restrict__ X,
                            const float* __restrict__ W,
                            float* __restrict__ Y) {
    __shared__ float Bs[CPAD * KSTRIDE];

    // cooperative transpose+pad fill: Bs[c*KSTRIDE + k] = W[k*HC + c]
    for (int i = threadIdx.x; i < CPAD * DIN; i += 256) {
        int k = i / CPAD, c = i % CPAD;            // consecutive tids -> coalesced W row read
        Bs[c * KSTRIDE + k] = (c < HC) ? W[k * HC + c] : 0.f;
    }
    __syncthreads();

    int rowTile = blockIdx.x * 8 + (threadIdx.x >> 5);
    if (rowTile < NN / 16) {                       // wave-uniform: EXEC full in WMMA region
        int lane  = threadIdx.x & 31;
        int lhalf = lane & 15;
        int hi2   = (lane >= 16) ? 2 : 0;          // hi lanes carry K+2, K+3
        int row0  = rowTile * 16;
        const float* xrow = X + (long long)(row0 + lhalf) * DIN;

        v8f zero = {};
        v8f acc[CT];
        #pragma unroll
        for (int t = 0; t < CT; ++t) acc[t] = zero;

        for (int k = 0; k < DIN; k += 4) {
            int ka = k + hi2;
            v2f a = *(const v2f*)(xrow + ka);
            #pragma unroll
            for (int t = 0; t < CT; ++t) {
                v2f b = *(const v2f*)(&Bs[(t * 16 + lhalf) * KSTRIDE + ka]);
                acc[t] = __builtin_amdgcn_wmma_f32_16x16x4_f32(
                             false, a, false, b, (short)0, acc[t], false, false);
            }
        }

        int rbase = row0 + ((lane >= 16) ? 8 : 0); // D: VGPR r -> M=r / M=r+8
        #pragma unroll
        for (int t = 0; t < CT; ++t) {
            int bcol = t * 16 + lhalf;
            if (bcol < HC) {
                #pragma unroll
                for (int r = 0; r < 8; ++r)
                    Y[(long long)(rbase + r) * HC + bcol] = acc[t][r];
            }
        }
    }
}

// ---------------------------------------------------------------------------
// K3: a_src[n,h] = hs[n]·att_src[h];  a_dst[n,h] = x_t[n]·w_eff[:,h]
// ---------------------------------------------------------------------------
__global__ void k_scores(const float* __restrict__ hs,
                         const float* __restrict__ xt,
                         const float* __restrict__ att_src,
                         const float* __restrict__ weff,
                         float* __restrict__ a_src,
                         float* __restrict__ a_dst) {
    int n = blockIdx.x * blockDim.x + threadIdx.x;
    if (n >= NN) return;
    const float* h = hs + (long long)n * HC;
    float s0 = 0.f, s1 = 0.f;
    #pragma unroll 10
    for (int c = 0; c < CC; ++c) {
        s0 += h[c]      * att_src[c];
        s1 += h[CC + c] * att_src[CC + c];
    }
    a_src[n * HH + 0] = s0;
    a_src[n * HH + 1] = s1;

    const float* x = xt + (long long)n * DIN;
    float d0 = 0.f, d1 = 0.f;
    #pragma unroll 8
    for (int d = 0; d < DIN; ++d) {
        float v = x[d];
        d0 += v * weff[d * HH + 0];
        d1 += v * weff[d * HH + 1];
    }
    a_dst[n * HH + 0] = d0;
    a_dst[n * HH + 1] = d1;
}

// ---------------------------------------------------------------------------
// K4: e = leaky_relu(a_src[src]+a_dst[dst]); segment max (ordered-int atomic)
// ---------------------------------------------------------------------------
__global__ void k_edge_max(const int* __restrict__ src, const int* __restrict__ dst,
                           const float* __restrict__ a_src, const float* __restrict__ a_dst,
                           float* __restrict__ ex, int* __restrict__ mI) {
    int t = blockIdx.x * blockDim.x + threadIdx.x;
    if (t >= EE * HH) return;
    int e = t >> 1, h = t & 1;
    int s = src[e], d = dst[e];
    float v = a_src[s * HH + h] + a_dst[d * HH + h];
    v = (v > 0.f) ? v : NEG * v;
    ex[t] = v;                                    // ex layout == [e*2+h]
    atomicMax(&mI[d * HH + h], f2key(v));
}

// ---------------------------------------------------------------------------
// K5: ex = exp(e - m[dst]); denom[dst] += ex
// ---------------------------------------------------------------------------
__global__ void k_edge_exp(const int* __restrict__ dst,
                           const int* __restrict__ mI,
                           float* __restrict__ ex,
                           float* __restrict__ denom) {
    int t = blockIdx.x * blockDim.x + threadIdx.x;
    if (t >= EE * HH) return;
    int e = t >> 1, h = t & 1;
    int d = dst[e];
    float m = key2f(mI[d * HH + h]);
    float r = __expf(ex[t] - m);
    ex[t] = r;
    atomicAdd(&denom[d * HH + h], r);
}

// ---------------------------------------------------------------------------
// K6: out[dst] += alpha * hs[src]  — one wave32 per edge, lanes over channels
// ---------------------------------------------------------------------------
__global__ void k_edge_scatter(const int* __restrict__ src, const int* __restrict__ dst,
                               const float* __restrict__ ex, const float* __restrict__ denom,
                               const float* __restrict__ hs, float* __restrict__ outf) {
    int wave = blockIdx.x * (blockDim.x >> 5) + (threadIdx.x >> 5);
    if (wave >= EE) return;
    int lane = threadIdx.x & 31;
    int s = src[wave], d = dst[wave];
    float a0 = ex[wave * HH + 0] / (denom[d * HH + 0] + 1e-16f);
    float a1 = ex[wave * HH + 1] / (denom[d * HH + 1] + 1e-16f);
    const float* hrow = hs + (long long)s * HC;
    float* orow = outf + (long long)d * HC;
    for (int j = lane; j < HC; j += 32) {
        float al = (j < CC) ? a0 : a1;
        atomicAdd(&orow[j], al * hrow[j]);
    }
}

// ---------------------------------------------------------------------------
// K7: x = relu(out + bias); pooled[batch[n]] += x; counts[batch[n]] += 1
// ---------------------------------------------------------------------------
__global__ void k_node_pool(const float* __restrict__ outf,
                            const float* __restrict__ bias,
                            const int* __restrict__ batch,
                            float* __restrict__ pooled,
                            float* __restrict__ counts) {
    long long t = (long long)blockIdx.x * blockDim.x + threadIdx.x;
    if (t >= (long long)NN * HC) return;
    int n = (int)(t / HC), j = (int)(t % HC);
    float v = outf[t] + bias[j];
    v = (v > 0.f) ? v : 0.f;
    int b = batch[n];
    atomicAdd(&pooled[b * HC + j], v);
    if (j == 0) atomicAdd(&counts[b], 1.0f);
}

// ---------------------------------------------------------------------------
// K8: y[b] = (pooled[b]/max(counts,1)) @ fc_w + fc_b
// ---------------------------------------------------------------------------
__global__ void k_fc(const float* __restrict__ pooled, const float* __restrict__ counts,
                     const float* __restrict__ fc_w, const float* __restrict__ fc_b,
                     float* __restrict__ y) {
    int b = threadIdx.x;
    if (b >= BB) return;
    float cnt = counts[b];
    cnt = (cnt > 1.0f) ? cnt : 1.0f;
    float inv = 1.0f / cnt;
    float acc = fc_b[0];
    #pragma unroll 10
    for (int c = 0; c < HC; ++c)
        acc += pooled[b * HC + c] * inv * fc_w[c];
    y[b] = acc;
}

// ---------------------------------------------------------------------------
extern "C" void kernel_launch(void* const* d_in, const int* in_sizes, int n_in,
                              void* d_out, int out_size, void* d_ws, size_t ws_size,
                              hipStream_t stream) {
    const float* x_s      = (const float*)d_in[0];
    const float* x_t      = (const float*)d_in[1];
    const int*   eidx     = (const int*)d_in[2];     // [2, E]
    // d_in[3] distances: unused (edge_dim=None)
    const int*   xs_batch = (const int*)d_in[4];
    // d_in[5] x_t_batch: unused
    const float* W_src    = (const float*)d_in[6];
    const float* W_dst    = (const float*)d_in[7];
    const float* att_src  = (const float*)d_in[8];
    const float* att_dst  = (const float*)d_in[9];
    const float* bias     = (const float*)d_in[10];
    const float* fc_w     = (const float*)d_in[11];
    const float* fc_b     = (const float*)d_in[12];
    float*       y        = (float*)d_out;

    const int* src = eidx;
    const int* dst = eidx + EE;

    // ---- carve workspace (≈96 MB) ----
    char* p = (char*)d_ws;
    size_t off = 0;
    auto take = [&](size_t bytes) {
        char* q = p + off;
        off = (off + bytes + 255) & ~(size_t)255;
        return q;
    };
    float* hs     = (float*)take((size_t)NN * HC * 4);
    float* outf   = (float*)take((size_t)NN * HC * 4);
    float* a_src  = (float*)take((size_t)NN * HH * 4);
    float* a_dst  = (float*)take((size_t)NN * HH * 4);
    int*   mI     = (int*)  take((size_t)NN * HH * 4);
    float* denom  = (float*)take((size_t)NN * HH * 4);
    float* ex     = (float*)take((size_t)EE * HH * 4);
    float* pooled = (float*)take((size_t)BB * HC * 4);
    float* counts = (float*)take((size_t)BB * 4);
    float* weff   = (float*)take((size_t)DIN * HH * 4);
    (void)ws_size; (void)in_sizes; (void)n_in; (void)out_size;

    // K0: effective dst-attention matrix (128x2)
    k_prep<<<1, 256, 0, stream>>>(W_dst, att_dst, weff);

    // K1: init accumulators
    {
        long long total = (long long)NN * HC;
        int blocks = (int)((total + 255) / 256);
        k_init<<<blocks, 256, 0, stream>>>(outf, mI, denom, pooled, counts);
    }

    // K2: WMMA GEMM  hs = x_s @ W_src   (8 row-strips of 16 per block)
    {
        int blocks = (NN / 16 + 7) / 8;                  // 782
        k_gemm_wmma<<<blocks, 256, 0, stream>>>(x_s, W_src, hs);
    }

    // K3: attention scores
    k_scores<<<(NN + 255) / 256, 256, 0, stream>>>(hs, x_t, att_src, weff,
                                                   a_src, a_dst);

    // K4: leaky-relu + segment max
    k_edge_max<<<(EE * HH + 255) / 256, 256, 0, stream>>>(src, dst, a_src, a_dst,
                                                          ex, mI);

    // K5: exp + segment sum
    k_edge_exp<<<(EE * HH + 255) / 256, 256, 0, stream>>>(dst, mI, ex, denom);

    // K6: weighted scatter aggregation (wave per edge)
    k_edge_scatter<<<(EE + 7) / 8, 256, 0, stream>>>(src, dst, ex, denom, hs, outf);

    // K7: relu + bias + mean-pool numerators
    {
        long long total = (long long)NN * HC;
        int blocks = (int)((total + 255) / 256);
        k_node_pool<<<blocks, 256, 0, stream>>>(outf, bias, xs_batch, pooled, counts);
    }

    // K8: final fc -> y[64]
    k_fc<<<1, 64, 0, stream>>>(pooled, counts, fc_w, fc_b, y);
}